// ConstrainedAttentionModel_30107720745624
// MI455X (gfx1250) — compile-verified
//
#include <hip/hip_runtime.h>

// Problem constants (from reference setup_inputs)
#define T_LEN   16384
#define KDIM    8
#define VOCAB   32000
#define NEG_INF_F (-1000000000.0f)
#define NTHREADS 1024
#define NWAVES   (NTHREADS / 32)
#define SC_PAD   32            // band-accumulate spill room past T (max +22 needed)

// LDS layout: sc at offset 0 so ds_add_f32 can use immediate offsets
#define SC_BYTES   ((T_LEN + SC_PAD) * 4)        // 65664
#define RED_OFF    SC_BYTES                      // 65664
#define XS_OFF     (RED_OFF + 256)               // 65920
#define SMEM_BYTES (XS_OFF + T_LEN * 4)          // 131456

typedef __attribute__((ext_vector_type(2))) float v2f;
typedef __attribute__((ext_vector_type(8))) float v8f;

// ---------------------------------------------------------------------------
// Kernel 1: zero the output histogram (harness poisons d_out; scatter is +=).
// ---------------------------------------------------------------------------
__global__ void zero_out_kernel(float* __restrict__ out, int n) {
    for (int i = blockIdx.x * blockDim.x + threadIdx.x; i < n;
         i += gridDim.x * blockDim.x)
        out[i] = 0.0f;
}

// ---------------------------------------------------------------------------
// Kernel 2: one workgroup per batch row (B=64 blocks x 1024 threads, 32 waves).
//   Phase A: stage x[b,:] (low 32 bits of int64 tokens) into LDS, zero scores.
//   Phase B: P = M @ C via V_WMMA_F32_16X16X4_F32 (two chained K=4 WMMAs per
//            16-row tile; scalar-uniform tile loop; prefetched token loads),
//            band-accumulate P[s,i] into scores[s+i] via ds_add_f32 with
//            immediate offsets into a padded score array at LDS offset 0.
//   Phase C: in-LDS softmax over T (wave32 shuffle + LDS reductions).
//   Phase D: global_atomic_add_f32 scatter into out[b, token].
// ---------------------------------------------------------------------------
__global__ __launch_bounds__(NTHREADS)
void constrained_attn_kernel(const float* __restrict__ C,
                             const long long* __restrict__ x,
                             float* __restrict__ out) {
    extern __shared__ char smem[];
    float* sc  = (float*)smem;                  // offset 0: scores / exp (hot)
    float* red = (float*)(smem + RED_OFF);      // reduction scratch
    int*   xs  = (int*)(smem + XS_OFF);         // token row as int32

    const int b    = blockIdx.x;
    const int tid  = threadIdx.x;
    const int lane = tid & 31;
    const int wave = tid >> 5;
    const int half = lane >> 4;   // which 16-lane half of the wave
    const int n16  = lane & 15;

    const long long* xrow = x + (long long)b * T_LEN;

    // ---- Phase A: stage tokens (values < 32000 -> low word of int64) ----
    {
        const longlong2* xr2 = (const longlong2*)xrow;
        for (int t = 2 * tid; t < T_LEN; t += 2 * NTHREADS) {
            longlong2 v = xr2[t >> 1];
            xs[t]     = (int)v.x;
            xs[t + 1] = (int)v.y;
            sc[t]     = 0.0f;
            sc[t + 1] = 0.0f;
        }
    }
    __syncthreads();

    // q_tok[j] = x[b, T-1-j]
    int q[KDIM];
#pragma unroll
    for (int j = 0; j < KDIM; ++j) q[j] = xs[T_LEN - 1 - j];

    // ---- B fragments: B[k][n] = C[j=k][i=n], cols n>=8 padded with zero ----
    // Layout (4x16 f32 B): lane n16 = column; K = 2*half + vgpr.
    v2f B0, B1;
    B0.x = (n16 < KDIM) ? C[(2 * half + 0) * KDIM + n16] : 0.0f;
    B0.y = (n16 < KDIM) ? C[(2 * half + 1) * KDIM + n16] : 0.0f;
    B1.x = (n16 < KDIM) ? C[(2 * half + 4) * KDIM + n16] : 0.0f;
    B1.y = (n16 < KDIM) ? C[(2 * half + 5) * KDIM + n16] : 0.0f;

    // ---- Phase B: scalar-uniform loop over 16-row tiles; prefetched token ----
    const int NITER = T_LEN / (16 * NWAVES);   // 32
    int tok = xs[wave * 16 + n16];             // tile 0 token (A row = n16)
    for (int it = 0; it < NITER; ++it) {
        const int s0 = (it * NWAVES + wave) * 16;

        // A layout (16x4 f32): lane row m = n16; K = 2*half + vgpr.
        v2f A0, A1;
        A0.x = (tok == q[2 * half + 0]) ? 1.0f : 0.0f;
        A0.y = (tok == q[2 * half + 1]) ? 1.0f : 0.0f;
        A1.x = (tok == q[2 * half + 4]) ? 1.0f : 0.0f;
        A1.y = (tok == q[2 * half + 5]) ? 1.0f : 0.0f;

        // branchless prefetch of next tile's token (wrapped index stays in range)
        tok = xs[((s0 + NWAVES * 16) & (T_LEN - 1)) + n16];

        v8f acc = {};
        // D[r][i] = sum_j M[s0+r, j] * C[j][i]  (j split 0..3 / 4..7)
        acc = __builtin_amdgcn_wmma_f32_16x16x4_f32(
                  false, A0, false, B0, (short)0, acc, false, false);
        acc = __builtin_amdgcn_wmma_f32_16x16x4_f32(
                  false, A1, false, B1, (short)0, acc, false, false);

        // D layout: lane col i = n16, rows r = v + 8*half.
        // scores[s+i] += P[s,i]; spill past T lands in SC_PAD (never read).
        // sc is at LDS offset 0 -> the 8 adds use immediate offsets 0..28.
        if (n16 < KDIM) {
            float* dst = &sc[s0 + 8 * half + n16];
#pragma unroll
            for (int v = 0; v < 8; ++v)
                atomicAdd(&dst[v], acc[v]);
        }
    }
    __syncthreads();

    if (tid == 0) sc[T_LEN - 1] = NEG_INF_F;
    __syncthreads();

    // ---- Phase C: softmax over sc[0..T) ----
    float mx = -3.4e38f;
    for (int t = tid; t < T_LEN; t += NTHREADS) mx = fmaxf(mx, sc[t]);
#pragma unroll
    for (int o = 16; o > 0; o >>= 1) mx = fmaxf(mx, __shfl_xor(mx, o, 32));
    if (lane == 0) red[wave] = mx;
    __syncthreads();
    if (tid == 0) {
        float m = red[0];
        for (int w = 1; w < NWAVES; ++w) m = fmaxf(m, red[w]);
        red[NWAVES] = m;
    }
    __syncthreads();
    mx = red[NWAVES];

    float sum = 0.0f;
    for (int t = tid; t < T_LEN; t += NTHREADS) {
        const float e = expf(sc[t] - mx);
        sc[t] = e;
        sum += e;
    }
#pragma unroll
    for (int o = 16; o > 0; o >>= 1) sum += __shfl_xor(sum, o, 32);
    if (lane == 0) red[wave] = sum;
    __syncthreads();
    if (tid == 0) {
        float s = 0.0f;
        for (int w = 0; w < NWAVES; ++w) s += red[w];
        red[NWAVES] = s;
    }
    __syncthreads();
    const float inv = 1.0f / red[NWAVES];

    // ---- Phase D: scatter attn into out[b, token] ----
    float* orow = out + (long long)b * VOCAB;
    for (int t = tid; t < T_LEN; t += NTHREADS) {
        atomicAdd(&orow[xs[t]], sc[t] * inv);
    }
}

// ---------------------------------------------------------------------------
extern "C" void kernel_launch(void* const* d_in, const int* in_sizes, int n_in,
                              void* d_out, int out_size, void* d_ws, size_t ws_size,
                              hipStream_t stream) {
    const float*     d_C = (const float*)d_in[0];
    const long long* d_x = (const long long*)d_in[1];
    float*           out = (float*)d_out;

    const int B = in_sizes[1] / T_LEN;   // 64

    // zero output (scatter accumulates; harness poisons d_out)
    {
        int blocks = (out_size + 1023) / 1024;
        if (blocks > 2048) blocks = 2048;
        zero_out_kernel<<<blocks, 1024, 0, stream>>>(out, out_size);
    }

    constrained_attn_kernel<<<B, NTHREADS, (size_t)SMEM_BYTES, stream>>>(d_C, d_x, out);
}